// FLAME_14096082665883
// MI455X (gfx1250) — compile-verified
//
#include <hip/hip_runtime.h>
#include <hip/hip_bf16.h>
#include <math.h>

// ---------------- problem constants ----------------
constexpr int B_  = 512;
constexpr int V_  = 5023;
constexpr int F_  = 9976;
constexpr int J_  = 5;
constexpr int S_  = 100;
constexpr int E_  = 50;
constexpr int K_  = S_ + E_;        // 150
constexpr int KP_ = 152;            // K padded to multiple of 4
constexpr int N_  = V_ * 3;         // 15069
constexpr int PK_ = 36;             // (J-1)*9 pose-feature K
constexpr int LSTAT_ = 51;
constexpr int LDYN_  = 17;
constexpr int LFULL_ = 68;
constexpr int L2D_   = LDYN_ + LSTAT_;   // 68

typedef float v2f __attribute__((ext_vector_type(2)));
typedef float v8f __attribute__((ext_vector_type(8)));

// workspace layout (floats)
constexpr int WS_JS   = 0;                       // [15][150]
constexpr int WS_JT   = WS_JS + 15 * K_;         // [15]
constexpr int WS_PF   = 2272;                    // [B][36]
constexpr int WS_AREL = WS_PF + B_ * PK_;        // [B][5][12]  (3 rows x {r0 r1 r2 t})
constexpr int WS_YROT = WS_AREL + B_ * J_ * 12;  // [B] ints

// output layout (floats)
constexpr size_t OUT_V  = 0;
constexpr size_t OUT_2D = (size_t)B_ * V_ * 3;
constexpr size_t OUT_3D = OUT_2D + (size_t)B_ * L2D_ * 3;

// =====================================================================
// Kernel A: JS[jc][l] = sum_v Jr[j,v]*shapedirs[v,c,l] ; Jt[jc] = Jr . vt
// grid (151, 15) x 256 ; x==150 -> template column
// =====================================================================
__global__ __launch_bounds__(256)
void js_kernel(const float* __restrict__ Jr, const float* __restrict__ SD,
               const float* __restrict__ VT, float* __restrict__ JS,
               float* __restrict__ Jt) {
    __shared__ float red[256];
    const int l  = blockIdx.x;          // 0..150
    const int jc = blockIdx.y;          // 0..14
    const int j = jc / 3, c = jc % 3;
    float acc = 0.f;
    if (l < K_) {
        for (int v = threadIdx.x; v < V_; v += 256)
            acc += Jr[j * V_ + v] * SD[(size_t)(v * 3 + c) * K_ + l];
    } else {
        for (int v = threadIdx.x; v < V_; v += 256)
            acc += Jr[j * V_ + v] * VT[v * 3 + c];
    }
    red[threadIdx.x] = acc;
    __syncthreads();
    for (int s = 128; s > 0; s >>= 1) {
        if (threadIdx.x < s) red[threadIdx.x] += red[threadIdx.x + s];
        __syncthreads();
    }
    if (threadIdx.x == 0) {
        if (l < K_) JS[jc * K_ + l] = red[0];
        else        Jt[jc]          = red[0];
    }
}

// =====================================================================
// Kernel B: per-batch Rodrigues, joints, kinematic chain, A_rel, y_rot
// =====================================================================
__device__ inline void rodrigues3(const float r0, const float r1, const float r2,
                                  float R[9]) {
    float ox = r0 + 1e-8f, oy = r1 + 1e-8f, oz = r2 + 1e-8f;
    float angle = sqrtf(ox * ox + oy * oy + oz * oz);
    float inv = 1.0f / angle;
    float ax = r0 * inv, ay = r1 * inv, az = r2 * inv;
    float s = sinf(angle), c = cosf(angle), t = 1.0f - c;
    float aa = ax * ax + ay * ay + az * az;   // K^2 = a a^T - (a.a) I
    R[0] = 1.f + t * (ax * ax - aa); R[1] = -s * az + t * (ax * ay); R[2] =  s * ay + t * (ax * az);
    R[3] =  s * az + t * (ay * ax);  R[4] = 1.f + t * (ay * ay - aa); R[5] = -s * ax + t * (ay * az);
    R[6] = -s * ay + t * (az * ax);  R[7] =  s * ax + t * (az * ay);  R[8] = 1.f + t * (az * az - aa);
}

__global__ __launch_bounds__(256)
void batch_kernel(const float* __restrict__ shp, const float* __restrict__ expr,
                  const float* __restrict__ pose, const float* __restrict__ eye,
                  const float* __restrict__ JS, const float* __restrict__ Jt,
                  float* __restrict__ pf_out, float* __restrict__ arel_out,
                  int* __restrict__ yrot_out) {
    const int b = blockIdx.x * blockDim.x + threadIdx.x;
    if (b >= B_) return;

    float fp[15];
    fp[0] = pose[b * 6 + 0]; fp[1] = pose[b * 6 + 1]; fp[2] = pose[b * 6 + 2];
    fp[3] = 0.f; fp[4] = 0.f; fp[5] = 0.f;
    fp[6] = pose[b * 6 + 3]; fp[7] = pose[b * 6 + 4]; fp[8] = pose[b * 6 + 5];
    fp[9]  = eye[b * 6 + 0]; fp[10] = eye[b * 6 + 1]; fp[11] = eye[b * 6 + 2];
    fp[12] = eye[b * 6 + 3]; fp[13] = eye[b * 6 + 4]; fp[14] = eye[b * 6 + 5];

    float R[5][9];
    #pragma unroll
    for (int j = 0; j < J_; ++j)
        rodrigues3(fp[j * 3 + 0], fp[j * 3 + 1], fp[j * 3 + 2], R[j]);

    // pose_feature = (R[1:] - I).flatten
    #pragma unroll
    for (int j = 1; j < J_; ++j)
        #pragma unroll
        for (int e = 0; e < 9; ++e)
            pf_out[b * PK_ + (j - 1) * 9 + e] = R[j][e] - ((e % 4 == 0) ? 1.f : 0.f);

    // joints = Jt + JS . betas
    float joints[5][3];
    for (int jc = 0; jc < 15; ++jc) {
        float a = Jt[jc];
        const float* js = JS + jc * K_;
        for (int l = 0; l < S_; ++l) a += js[l] * shp[b * S_ + l];
        for (int l = 0; l < E_; ++l) a += js[S_ + l] * expr[b * E_ + l];
        joints[jc / 3][jc % 3] = a;
    }

    const int parents[5] = {-1, 0, 1, 1, 1};
    float relJ[5][3];
    #pragma unroll
    for (int c = 0; c < 3; ++c) relJ[0][c] = joints[0][c];
    for (int j = 1; j < J_; ++j)
        #pragma unroll
        for (int c = 0; c < 3; ++c) relJ[j][c] = joints[j][c] - joints[parents[j]][c];

    // kinematic chain (affine 3x4)
    float Gr[5][9], Gt[5][3];
    #pragma unroll
    for (int e = 0; e < 9; ++e) Gr[0][e] = R[0][e];
    #pragma unroll
    for (int c = 0; c < 3; ++c) Gt[0][c] = relJ[0][c];
    for (int j = 1; j < J_; ++j) {
        const int p = parents[j];
        for (int r = 0; r < 3; ++r)
            for (int c = 0; c < 3; ++c) {
                float a = 0.f;
                for (int k = 0; k < 3; ++k) a += Gr[p][r * 3 + k] * R[j][k * 3 + c];
                Gr[j][r * 3 + c] = a;
            }
        for (int r = 0; r < 3; ++r) {
            float a = Gt[p][r];
            for (int k = 0; k < 3; ++k) a += Gr[p][r * 3 + k] * relJ[j][k];
            Gt[j][r] = a;
        }
    }

    // A_rel: rot = Gr, trans = Gt - Gr*joints
    for (int j = 0; j < J_; ++j) {
        for (int c = 0; c < 3; ++c) {
            float rj = 0.f;
            for (int k = 0; k < 3; ++k) rj += Gr[j][c * 3 + k] * joints[j][k];
            float* o = arel_out + (size_t)b * 60 + j * 12 + c * 4;
            o[0] = Gr[j][c * 3 + 0];
            o[1] = Gr[j][c * 3 + 1];
            o[2] = Gr[j][c * 3 + 2];
            o[3] = Gt[j][c] - rj;
        }
    }

    // neck chain: rel_rot = R[0] @ R[1]; need column 0
    float M[3];
    #pragma unroll
    for (int c = 0; c < 3; ++c)
        M[c] = R[0][c * 3 + 0] * R[1][0] + R[0][c * 3 + 1] * R[1][3] + R[0][c * 3 + 2] * R[1][6];
    float sy = sqrtf(M[0] * M[0] + M[1] * M[1]);
    float ydeg = atan2f(-M[2], sy) * 57.29577951308232f;
    int yr = (int)rintf(fminf(ydeg, 39.0f));
    if (yr < 0) yr = (yr < -39) ? 78 : (39 - yr);
    yrot_out[b] = yr;
}

// =====================================================================
// Kernel C: fused WMMA GEMMs + LBS skinning.
//   block = 128 thr = 4 waves; wave -> 16 batch rows x 16 vertices (48 cols)
//   grid  = (ceil(V/64), B/16)
//   Inner loops are branchless: out-of-range columns are clamped (their
//   garbage results map to vertices >= V and are never read), and the
//   K tail (150..151) multiplies zero-padded A rows so B may read the
//   valid K=148..149 floats instead.
// =====================================================================
constexpr int AST = 154;   // betas LDS row stride (even -> b64 aligned; 154%64=26, gcd 2 -> conflict-free for 16 lanes)
constexpr int PST = 38;    // pose-feature LDS row stride
constexpr int VST = 49;    // v_posed tile row stride

__global__ __launch_bounds__(128)
void lbs_kernel(const float* __restrict__ shp, const float* __restrict__ expr,
                const float* __restrict__ VT, const float* __restrict__ SD,
                const float* __restrict__ PD, const float* __restrict__ LW,
                const float* __restrict__ pf_ws, const float* __restrict__ arel_ws,
                float* __restrict__ vout) {
    __shared__ float Ab[16 * AST];       // betas tile   [m][k], zero padded K 150..151
    __shared__ float PFs[16 * PST];      // pose feat    [m][k]
    __shared__ float Ars[16 * 60];       // A_rel        [m][j*12 + e]
    __shared__ float Pl[4 * 16 * VST];   // v_posed tile per wave [m][48]

    const int tid  = threadIdx.x;
    const int lane = tid & 31;
    const int wave = tid >> 5;
    const int half = lane >> 4;
    const int col  = lane & 15;
    const int m0   = blockIdx.y * 16;
    const int vbase = blockIdx.x * 64 + wave * 16;

    // ---- stage per-M-tile operands ----
    for (int idx = tid; idx < 16 * KP_; idx += 128) {
        int m = idx / KP_, k = idx % KP_;
        float v;
        if (k < S_)       v = shp[(m0 + m) * S_ + k];
        else if (k < K_)  v = expr[(m0 + m) * E_ + (k - S_)];
        else              v = 0.f;
        Ab[m * AST + k] = v;
    }
    for (int idx = tid; idx < 16 * PK_; idx += 128) {
        int m = idx / PK_, k = idx % PK_;
        PFs[m * PST + k] = pf_ws[(m0 + m) * PK_ + k];
    }
    for (int idx = tid; idx < 16 * 60; idx += 128) {
        int m = idx / 60, e = idx % 60;
        Ars[m * 60 + e] = arel_ws[(size_t)(m0 + m) * 60 + e];
    }
    __syncthreads();

    float* Pw = &Pl[wave * 16 * VST];
    const float* Arow = &Ab[col * AST + 2 * half];

    // ---- 3 WMMA n-tiles per wave: v_posed = betas*SD^T + pf*PD + vt ----
    for (int s = 0; s < 3; ++s) {
        const int n  = vbase * 3 + s * 16 + col;
        const int nc = (n < N_) ? n : (N_ - 1);          // clamp; OOB cols discarded later
        const float* sdrow = SD + (size_t)nc * K_;        // row of shapedirs (len 150)
        const float* sdh   = sdrow + 2 * half;            // per-half K base
        const float* pdh   = PD + (size_t)(2 * half) * N_ + nc;
        v8f acc = {0.f, 0.f, 0.f, 0.f, 0.f, 0.f, 0.f, 0.f};

        // K = 0..147 : 37 unconditional steps, b64 loads with immediate offsets
        #pragma unroll 4
        for (int k = 0; k < 148; k += 4) {
            v2f a = *(const v2f*)(Arow + k);
            v2f b = *(const v2f*)(sdh + k);
            acc = __builtin_amdgcn_wmma_f32_16x16x4_f32(
                false, a, false, b, (short)0, acc, false, false);
        }
        // K tail 148..151: A rows 150,151 are zero, so both halves may read
        // the valid floats at K=148,149 for the don't-care positions.
        {
            v2f a = *(const v2f*)(Arow + 148);            // half1 -> zeros
            v2f b = *(const v2f*)(sdrow + 148);
            acc = __builtin_amdgcn_wmma_f32_16x16x4_f32(
                false, a, false, b, (short)0, acc, false, false);
        }
        // pose correctives: K = 36, unconditional
        const float* PFrow = &PFs[col * PST + 2 * half];
        #pragma unroll
        for (int k = 0; k < PK_; k += 4) {
            v2f a = *(const v2f*)(PFrow + k);
            v2f b;
            b.x = pdh[(size_t)k * N_];
            b.y = pdh[(size_t)(k + 1) * N_];
            acc = __builtin_amdgcn_wmma_f32_16x16x4_f32(
                false, a, false, b, (short)0, acc, false, false);
        }
        const float vtv = VT[nc];
        #pragma unroll
        for (int r = 0; r < 8; ++r)
            Pw[(r + 8 * half) * VST + s * 16 + col] = acc[r] + vtv;
    }
    __syncthreads();

    // ---- skinning: 16 m x 16 v per wave, 8 items/lane ----
    #pragma unroll
    for (int i = 0; i < 8; ++i) {
        const int it = lane + i * 32;
        const int vl = it & 15;
        const int m  = it >> 4;
        const int v  = vbase + vl;
        if (v < V_) {
            const float px = Pw[m * VST + vl * 3 + 0];
            const float py = Pw[m * VST + vl * 3 + 1];
            const float pz = Pw[m * VST + vl * 3 + 2];
            float w[5];
            #pragma unroll
            for (int j = 0; j < J_; ++j) w[j] = LW[(size_t)v * J_ + j];
            const float* Am = &Ars[m * 60];
            #pragma unroll
            for (int c = 0; c < 3; ++c) {
                float t0 = 0.f, t1 = 0.f, t2 = 0.f, t3 = 0.f;
                #pragma unroll
                for (int j = 0; j < J_; ++j) {
                    const float* a = Am + j * 12 + c * 4;
                    t0 += w[j] * a[0]; t1 += w[j] * a[1];
                    t2 += w[j] * a[2]; t3 += w[j] * a[3];
                }
                vout[((size_t)(m0 + m) * V_ + v) * 3 + c] =
                    t0 * px + t1 * py + t2 * pz + t3;
            }
        }
    }
}

// =====================================================================
// Kernel D: landmark barycentric gathers (2D dynamic+static, 3D full)
// =====================================================================
__global__ __launch_bounds__(256)
void lmk_kernel(const float* __restrict__ verts, const int* __restrict__ faces,
                const int* __restrict__ lmkF, const float* __restrict__ lmkB,
                const int* __restrict__ dynF, const float* __restrict__ dynB,
                const int* __restrict__ fullF, const float* __restrict__ fullB,
                const int* __restrict__ yrot,
                float* __restrict__ out2d, float* __restrict__ out3d) {
    const int idx = blockIdx.x * blockDim.x + threadIdx.x;
    if (idx >= B_ * (L2D_ + LFULL_)) return;
    const int b = idx / (L2D_ + LFULL_);
    const int l = idx % (L2D_ + LFULL_);

    int face; const float* bw; float* o;
    if (l < L2D_) {
        if (l < LDYN_) {
            const int yb = yrot[b];
            face = dynF[yb * LDYN_ + l];
            bw   = dynB + ((size_t)yb * LDYN_ + l) * 3;
        } else {
            face = lmkF[l - LDYN_];
            bw   = lmkB + (size_t)(l - LDYN_) * 3;
        }
        o = out2d + ((size_t)b * L2D_ + l) * 3;
    } else {
        const int l2 = l - L2D_;
        face = fullF[l2];
        bw   = fullB + (size_t)l2 * 3;
        o = out3d + ((size_t)b * LFULL_ + l2) * 3;
    }

    float a0 = 0.f, a1 = 0.f, a2 = 0.f;
    #pragma unroll
    for (int k = 0; k < 3; ++k) {
        const int vi = faces[face * 3 + k];
        const float w = bw[k];
        const float* vp = verts + ((size_t)b * V_ + vi) * 3;
        a0 += w * vp[0]; a1 += w * vp[1]; a2 += w * vp[2];
    }
    o[0] = a0; o[1] = a1; o[2] = a2;
}

// =====================================================================
extern "C" void kernel_launch(void* const* d_in, const int* in_sizes, int n_in,
                              void* d_out, int out_size, void* d_ws, size_t ws_size,
                              hipStream_t stream) {
    (void)in_sizes; (void)n_in; (void)out_size; (void)ws_size;
    const float* shp   = (const float*)d_in[0];
    const float* expr  = (const float*)d_in[1];
    const float* pose  = (const float*)d_in[2];
    const float* eye   = (const float*)d_in[3];
    const float* vt    = (const float*)d_in[4];
    const float* sd    = (const float*)d_in[5];
    const float* pd    = (const float*)d_in[6];
    const float* jr    = (const float*)d_in[7];
    const float* lw    = (const float*)d_in[8];
    const int*   faces = (const int*)d_in[9];
    const int*   lmkF  = (const int*)d_in[10];
    const float* lmkB  = (const float*)d_in[11];
    const int*   dynF  = (const int*)d_in[12];
    const float* dynB  = (const float*)d_in[13];
    const int*   fullF = (const int*)d_in[14];
    const float* fullB = (const float*)d_in[15];

    float* ws   = (float*)d_ws;
    float* JS   = ws + WS_JS;
    float* Jt   = ws + WS_JT;
    float* pf   = ws + WS_PF;
    float* arel = ws + WS_AREL;
    int*   yrot = (int*)(ws + WS_YROT);

    float* outV  = (float*)d_out + OUT_V;
    float* out2d = (float*)d_out + OUT_2D;
    float* out3d = (float*)d_out + OUT_3D;

    // A: fold J_regressor into shapedirs / template
    js_kernel<<<dim3(K_ + 1, 15), 256, 0, stream>>>(jr, sd, vt, JS, Jt);
    // B: per-batch pose pipeline
    batch_kernel<<<dim3((B_ + 255) / 256), 256, 0, stream>>>(
        shp, expr, pose, eye, JS, Jt, pf, arel, yrot);
    // C: fused WMMA blendshapes + skinning -> vertices
    lbs_kernel<<<dim3((V_ + 63) / 64, B_ / 16), 128, 0, stream>>>(
        shp, expr, vt, sd, pd, lw, pf, arel, outV);
    // D: landmarks
    const int nl = B_ * (L2D_ + LFULL_);
    lmk_kernel<<<dim3((nl + 255) / 256), 256, 0, stream>>>(
        outV, faces, lmkF, lmkB, dynF, dynB, fullF, fullB, yrot, out2d, out3d);
}